// BicliqueGCN_50431505989725
// MI455X (gfx1250) — compile-verified
//
#include <hip/hip_runtime.h>
#include <hip/hip_bf16.h>
#include <math.h>

#define N_NODES 50000
#define N_EDGES 800000
#define DIM 128

typedef __attribute__((ext_vector_type(2))) float v2f;
typedef __attribute__((ext_vector_type(8))) float v8f;

// ---- monotonic float <-> ordered-uint mapping (for atomic segment-max) ----
__device__ __forceinline__ unsigned f2ord(float f) {
  unsigned u = __float_as_uint(f);
  return (u & 0x80000000u) ? ~u : (u | 0x80000000u);
}
__device__ __forceinline__ float ord2f(unsigned u) {
  return __uint_as_float((u & 0x80000000u) ? (u & 0x7FFFFFFFu) : ~u);
}

// ---------------- degree / norm ----------------
__global__ void k_degree(const int* __restrict__ src, const int* __restrict__ dst,
                         float* __restrict__ degs, float* __restrict__ degd, int E) {
  int e = blockIdx.x * blockDim.x + threadIdx.x;
  if (e >= E) return;
  unsafeAtomicAdd(degs + src[e], 1.0f);
  unsafeAtomicAdd(degd + dst[e], 1.0f);
}

__global__ void k_invsqrt(float* __restrict__ a, int n) {
  int i = blockIdx.x * blockDim.x + threadIdx.x;
  if (i >= n) return;
  float d = a[i];
  a[i] = d > 0.0f ? rsqrtf(d) : 0.0f;
}

// ---------------- weight prep: transpose (+ optional per-K mask fold) ----------------
// Wt[n*128 + k] = W[k*128 + n] * (mask ? mask[k] : 1)
__global__ void k_prepW(const float* __restrict__ W, const float* __restrict__ mask,
                        float* __restrict__ Wt) {
  int i = blockIdx.x * blockDim.x + threadIdx.x;   // 0 .. 128*128-1
  int k = i >> 7, n = i & 127;
  float m = mask ? mask[k] : 1.0f;
  Wt[n * DIM + k] = W[i] * m;
}

// ---------------- fp32 WMMA GEMM: out[M,128] = A[M,128] @ W[128,128] ----------------
// Wt is the COLUMN-MAJOR weight (Wt[n*128+k] = W[k][n]); optional per-row scale.
template <bool HAS_SCALE>
__global__ __launch_bounds__(256)
void k_gemm_wmma(const float* __restrict__ A, const float* __restrict__ Wt,
                 float* __restrict__ out, const float* __restrict__ rowscale, int M) {
  __shared__ float Alds[16 * 132];          // 16 rows x 128 cols, padded stride
  const int tid  = threadIdx.x;
  const int wave = tid >> 5;
  const int lane = tid & 31;
  const int row0 = blockIdx.x * 16;
  if (row0 >= M) return;

  // cooperative load of 16x128 A strip (2 x float4 per thread, coalesced)
  for (int idx = tid; idx < 512; idx += 256) {
    int r  = idx >> 5;
    int c4 = (idx & 31) << 2;
    const float4 v = *(const float4*)(A + (size_t)(row0 + r) * DIM + c4);
    float* p = Alds + r * 132 + c4;
    p[0] = v.x; p[1] = v.y; p[2] = v.z; p[3] = v.w;
  }
  __syncthreads();

  const int hi   = lane >> 4;      // lane half selects K sub-pair {0,1} vs {2,3}
  const int l15  = lane & 15;
  const int col0 = wave * 16;      // 8 waves cover 128 output columns
  const float* arow = Alds + l15 * 132 + 2 * hi;                 // A frag base
  const float* wcol = Wt + (size_t)(col0 + l15) * DIM + 2 * hi;  // B frag base

  v8f c = {};
  #pragma unroll
  for (int kk = 0; kk < 32; ++kk) {
    // A fragment 16x4: lanes 0-15 K={4kk,4kk+1}, lanes 16-31 K={4kk+2,4kk+3}
    const v2f a = *(const v2f*)(arow + kk * 4);
    // B fragment 4x16: same K split; contiguous pair thanks to transposed W
    const v2f b = *(const v2f*)(wcol + kk * 4);
    c = __builtin_amdgcn_wmma_f32_16x16x4_f32(false, a, false, b, (short)0, c,
                                              false, false);
  }

  // C/D layout: VGPR r -> row (r + 8*hi), lane&15 -> column
  #pragma unroll
  for (int r = 0; r < 8; ++r) {
    int grow = row0 + r + 8 * hi;
    float sc = HAS_SCALE ? rowscale[grow] : 1.0f;
    out[(size_t)grow * DIM + col0 + l15] = c[r] * sc;
  }
}

// ---------------- edge scatter-add, one wave per edge, float4 per lane ----------------
__global__ __launch_bounds__(256)
void k_scatter(const int* __restrict__ src, const int* __restrict__ dst,
               const float* __restrict__ P, float* __restrict__ agg, int E) {
  int e    = (blockIdx.x * 256 + threadIdx.x) >> 5;
  int lane = threadIdx.x & 31;
  if (e >= E) return;
  const int s = src[e], d = dst[e];
  const float4 v = *(const float4*)(P + (size_t)s * DIM + lane * 4);
  float* q = agg + (size_t)d * DIM + lane * 4;
  unsafeAtomicAdd(q + 0, v.x);
  unsafeAtomicAdd(q + 1, v.y);
  unsafeAtomicAdd(q + 2, v.z);
  unsafeAtomicAdd(q + 3, v.w);
}

__global__ __launch_bounds__(256)
void k_scatter_attn(const int* __restrict__ src, const int* __restrict__ dst,
                    const float* __restrict__ P, const float* __restrict__ eedge,
                    const float* __restrict__ ssum, float* __restrict__ agg, int E) {
  int e    = (blockIdx.x * 256 + threadIdx.x) >> 5;
  int lane = threadIdx.x & 31;
  if (e >= E) return;
  const int s = src[e], d = dst[e];
  const float alpha = eedge[e] / ssum[d];
  const float4 v = *(const float4*)(P + (size_t)s * DIM + lane * 4);
  float* q = agg + (size_t)d * DIM + lane * 4;
  unsafeAtomicAdd(q + 0, v.x * alpha);
  unsafeAtomicAdd(q + 1, v.y * alpha);
  unsafeAtomicAdd(q + 2, v.z * alpha);
  unsafeAtomicAdd(q + 3, v.w * alpha);
}

// ---------------- per-node attention logit: score[i] = hm[i,:] . a ----------------
__global__ __launch_bounds__(256)
void k_rowdot(const float* __restrict__ hm, const float* __restrict__ a,
              float* __restrict__ score, int M) {
  int r    = (blockIdx.x * 256 + threadIdx.x) >> 5;
  int lane = threadIdx.x & 31;
  if (r >= M) return;
  const float4 h  = *(const float4*)(hm + (size_t)r * DIM + lane * 4);
  const float4 av = *(const float4*)(a + lane * 4);
  float s = h.x * av.x + h.y * av.y + h.z * av.z + h.w * av.w;
  #pragma unroll
  for (int off = 16; off > 0; off >>= 1) s += __shfl_xor(s, off, 32);
  if (lane == 0) score[r] = s;
}

// ---------------- edge softmax passes ----------------
__global__ void k_edge_max(const int* __restrict__ src, const int* __restrict__ dst,
                           const float* __restrict__ score, float* __restrict__ sedge,
                           unsigned* __restrict__ smax, int E) {
  int e = blockIdx.x * blockDim.x + threadIdx.x;
  if (e >= E) return;
  float s = score[src[e]];
  s = s > 0.0f ? s : 0.01f * s;           // leaky_relu, slope 0.01
  sedge[e] = s;
  atomicMax(smax + dst[e], f2ord(s));
}

__global__ void k_edge_exp(const int* __restrict__ dst, const unsigned* __restrict__ smax,
                           float* __restrict__ sedge /* in: s, out: exp */,
                           float* __restrict__ ssum, int E) {
  int e = blockIdx.x * blockDim.x + threadIdx.x;
  if (e >= E) return;
  const int d  = dst[e];
  const float ee = expf(sedge[e] - ord2f(smax[d]));
  sedge[e] = ee;
  unsafeAtomicAdd(ssum + d, ee);
}

// ---------------- pointwise ----------------
__global__ void k_norm_bias_relu(const float* __restrict__ in, const float* __restrict__ norm,
                                 const float* __restrict__ bias, float* __restrict__ out, int M) {
  int i = blockIdx.x * blockDim.x + threadIdx.x;
  if (i >= M * DIM) return;
  int row = i >> 7, col = i & 127;
  float v = in[i] * norm[row] + bias[col];
  out[i] = v > 0.0f ? v : 0.0f;
}

__global__ void k_relu(float* __restrict__ a, int n) {
  int i = blockIdx.x * blockDim.x + threadIdx.x;
  if (i >= n) return;
  float v = a[i];
  a[i] = v > 0.0f ? v : 0.0f;
}

extern "C" void kernel_launch(void* const* d_in, const int* in_sizes, int n_in,
                              void* d_out, int out_size, void* d_ws, size_t ws_size,
                              hipStream_t stream) {
  const int*   src  = (const int*)d_in[0];
  const int*   dst  = (const int*)d_in[1];
  const float* x    = (const float*)d_in[2];
  const float* mask = (const float*)d_in[3];
  const float* W1   = (const float*)d_in[4];
  const float* b1   = (const float*)d_in[5];
  const float* Wa   = (const float*)d_in[6];
  const float* av   = (const float*)d_in[7];
  const float* W2   = (const float*)d_in[8];
  const float* b2   = (const float*)d_in[9];
  float* out = (float*)d_out;

  char* ws = (char*)d_ws;
  auto carve = [&](size_t bytes) -> char* {
    char* p = ws;
    ws += (bytes + 255) & ~(size_t)255;
    return p;
  };
  float*    bufA     = (float*)carve((size_t)N_NODES * DIM * 4);
  float*    bufB     = (float*)carve((size_t)N_NODES * DIM * 4);
  float*    Wt1      = (float*)carve((size_t)DIM * DIM * 4);
  float*    WtA      = (float*)carve((size_t)DIM * DIM * 4);
  float*    Wt2      = (float*)carve((size_t)DIM * DIM * 4);
  float*    norm_src = (float*)carve((size_t)N_NODES * 4);
  float*    norm_dst = (float*)carve((size_t)N_NODES * 4);
  float*    score    = (float*)carve((size_t)N_NODES * 4);
  unsigned* smax     = (unsigned*)carve((size_t)N_NODES * 4);
  float*    ssum     = (float*)carve((size_t)N_NODES * 4);
  float*    eedge    = (float*)carve((size_t)N_EDGES * 4);

  const int TB = 256;
  dim3 bE((N_EDGES + TB - 1) / TB);
  dim3 bEw((N_EDGES * 32 + TB - 1) / TB);   // wave per edge
  dim3 bN((N_NODES + TB - 1) / TB);
  dim3 bNw((N_NODES * 32 + TB - 1) / TB);   // wave per node
  dim3 bND((N_NODES * DIM + TB - 1) / TB);
  dim3 bW((DIM * DIM) / TB);                // 64 blocks
  dim3 gGemm(N_NODES / 16);                 // 50000 = 3125*16, exact

  // weight prep: transposed (column-major) weights; mask folded into Wa
  k_prepW<<<bW, TB, 0, stream>>>(W1, nullptr, Wt1);
  k_prepW<<<bW, TB, 0, stream>>>(Wa, mask, WtA);
  k_prepW<<<bW, TB, 0, stream>>>(W2, nullptr, Wt2);

  // degrees -> symmetric norms (identical for both convs; compute once)
  hipMemsetAsync(norm_src, 0, (size_t)N_NODES * 4, stream);
  hipMemsetAsync(norm_dst, 0, (size_t)N_NODES * 4, stream);
  k_degree<<<bE, TB, 0, stream>>>(src, dst, norm_src, norm_dst, N_EDGES);
  k_invsqrt<<<bN, TB, 0, stream>>>(norm_src, N_NODES);
  k_invsqrt<<<bN, TB, 0, stream>>>(norm_dst, N_NODES);

  // ---- GraphConv 1 ----
  k_gemm_wmma<true><<<gGemm, TB, 0, stream>>>(x, Wt1, bufA, norm_src, N_NODES);
  hipMemsetAsync(bufB, 0, (size_t)N_NODES * DIM * 4, stream);
  k_scatter<<<bEw, TB, 0, stream>>>(src, dst, bufA, bufB, N_EDGES);
  k_norm_bias_relu<<<bND, TB, 0, stream>>>(bufB, norm_dst, b1, bufB, N_NODES);

  // ---- Biclique attention ----
  k_gemm_wmma<false><<<gGemm, TB, 0, stream>>>(bufB, WtA, bufA, nullptr, N_NODES); // hm
  k_rowdot<<<bNw, TB, 0, stream>>>(bufA, av, score, N_NODES);
  hipMemsetAsync(smax, 0, (size_t)N_NODES * 4, stream);  // 0 < f2ord(any float): valid -inf
  hipMemsetAsync(ssum, 0, (size_t)N_NODES * 4, stream);
  k_edge_max<<<bE, TB, 0, stream>>>(src, dst, score, eedge, smax, N_EDGES);
  k_edge_exp<<<bE, TB, 0, stream>>>(dst, smax, eedge, ssum, N_EDGES);
  hipMemsetAsync(bufB, 0, (size_t)N_NODES * DIM * 4, stream);
  k_scatter_attn<<<bEw, TB, 0, stream>>>(src, dst, bufA, eedge, ssum, bufB, N_EDGES);
  k_relu<<<bND, TB, 0, stream>>>(bufB, N_NODES * DIM);

  // ---- GraphConv 2 ----
  k_gemm_wmma<true><<<gGemm, TB, 0, stream>>>(bufB, Wt2, bufA, norm_src, N_NODES);
  hipMemsetAsync(bufB, 0, (size_t)N_NODES * DIM * 4, stream);
  k_scatter<<<bEw, TB, 0, stream>>>(src, dst, bufA, bufB, N_EDGES);
  k_norm_bias_relu<<<bND, TB, 0, stream>>>(bufB, norm_dst, b2, out, N_NODES);
}